// WordSAGE_3899830304782
// MI455X (gfx1250) — compile-verified
//
#include <hip/hip_runtime.h>
#include <hip/hip_bf16.h>

typedef __attribute__((ext_vector_type(2))) float v2f;
typedef __attribute__((ext_vector_type(8))) float v8f;

#define C128 128
#define LDSP 132   // padded LDS row stride (floats): 132 % 64 = 4 -> conflict-free,
                   // 132*4 = 528 bytes -> 16B aligned rows for float4

// ---------------- utility: zero fill ----------------
__global__ void fill_zero_kernel(float* __restrict__ p, long long n) {
    long long i = (long long)blockIdx.x * blockDim.x + threadIdx.x;
    long long stride = (long long)gridDim.x * blockDim.x;
    for (; i < n; i += stride) p[i] = 0.0f;
}

// ---------------- degree count ----------------
__global__ void degree_kernel(const int* __restrict__ dst, float* __restrict__ deg, int nedges) {
    int e = blockIdx.x * blockDim.x + threadIdx.x;
    if (e < nedges) atomicAdd(&deg[dst[e]], 1.0f);
}

__global__ void deginv_kernel(const float* __restrict__ deg, float* __restrict__ deg_inv, int n) {
    int i = blockIdx.x * blockDim.x + threadIdx.x;
    if (i < n) {
        float d = deg[i];
        deg_inv[i] = (d > 0.0f) ? (1.0f / d) : 0.0f;
    }
}

// ---------------- edge aggregation: one wave (32 lanes) per edge ----------------
__global__ void aggregate_kernel(const float* __restrict__ x,
                                 const int* __restrict__ src,
                                 const int* __restrict__ dst,
                                 float* __restrict__ agg, int nedges) {
    int wavesPerBlock = blockDim.x >> 5;
    int e = blockIdx.x * wavesPerBlock + (threadIdx.x >> 5);
    if (e >= nedges) return;
    int lane = threadIdx.x & 31;
    int s = src[e];
    int d = dst[e];
    const float4* xs = (const float4*)(x + (size_t)s * C128);
    float4 v = xs[lane];
    float* ad = agg + (size_t)d * C128 + lane * 4;
    atomicAdd(ad + 0, v.x);
    atomicAdd(ad + 1, v.y);
    atomicAdd(ad + 2, v.z);
    atomicAdd(ad + 3, v.w);
}

// ---------------- fused SAGE layer GEMM ----------------
// out[m,:] = relu( (agg[m,:]*deg_inv[m]) @ W1 + A2[m,:] @ W2 + bias )
// Block: 256 threads = 8 waves; 16-row tile per block; wave w owns cols [16w,16w+16).
// fp32 WMMA 16x16x4, K=128. Epilogue staged through LDS for coalesced b128 stores.
__global__ void __launch_bounds__(256)
sage_gemm_kernel(const float* __restrict__ Aagg,
                 const float* __restrict__ deg_inv,
                 const float* __restrict__ A2,
                 const float* __restrict__ W1,
                 const float* __restrict__ W2,
                 const float* __restrict__ bias,
                 float* __restrict__ out,
                 int nrows) {
    __shared__ float sA1[16][LDSP];   // deg_inv-scaled aggregate tile (reused as out-stage)
    __shared__ float sA2[16][LDSP];   // self-feature tile

    const int m0  = blockIdx.x * 16;
    const int tid = threadIdx.x;

    // cooperative float4 stage of both 16x128 tiles into LDS
    for (int idx = tid; idx < 16 * 32; idx += 256) {
        int row = idx >> 5;
        int c4  = idx & 31;
        int m = m0 + row;
        float4 a1 = {0.f, 0.f, 0.f, 0.f}, a2 = a1;
        if (m < nrows) {
            float di = deg_inv[m];
            float4 r1 = ((const float4*)(Aagg + (size_t)m * C128))[c4];
            a1.x = r1.x * di; a1.y = r1.y * di; a1.z = r1.z * di; a1.w = r1.w * di;
            a2 = ((const float4*)(A2 + (size_t)m * C128))[c4];
        }
        ((float4*)&sA1[row][0])[c4] = a1;
        ((float4*)&sA2[row][0])[c4] = a2;
    }
    __syncthreads();

    const int wave = tid >> 5;
    const int lane = tid & 31;
    const int n    = lane & 15;          // A: M index; B/C/D: N index within tile
    const int half = lane >> 4;          // K-half selector per ISA VGPR layout
    const int n0   = wave * 16;
    const int col  = n0 + n;

    v8f acc = {};
    for (int k0 = 0; k0 < C128; k0 += 4) {
        const int k = k0 + 2 * half;
        v2f a1, a2, b1, b2;
        a1.x = sA1[n][k];     a1.y = sA1[n][k + 1];
        a2.x = sA2[n][k];     a2.y = sA2[n][k + 1];
        b1.x = W1[(size_t)k * C128 + col];
        b1.y = W1[(size_t)(k + 1) * C128 + col];
        b2.x = W2[(size_t)k * C128 + col];
        b2.y = W2[(size_t)(k + 1) * C128 + col];
        acc = __builtin_amdgcn_wmma_f32_16x16x4_f32(false, a1, false, b1,
                                                    (short)0, acc, false, false);
        acc = __builtin_amdgcn_wmma_f32_16x16x4_f32(false, a2, false, b2,
                                                    (short)0, acc, false, false);
    }

    // stage D tile into LDS (reuse sA1), then coalesced write-out
    __syncthreads();
    #pragma unroll
    for (int i = 0; i < 8; ++i) {
        sA1[i + 8 * half][col] = acc[i];  // D layout: VGPR i -> M=i (lo lanes), M=8+i (hi lanes)
    }
    __syncthreads();

    for (int idx = tid; idx < 16 * 32; idx += 256) {
        int row = idx >> 5;
        int c4  = idx & 31;
        int m = m0 + row;
        if (m < nrows) {
            float4 v = ((const float4*)&sA1[row][0])[c4];
            float4 b = ((const float4*)bias)[c4];
            v.x = fmaxf(v.x + b.x, 0.0f);
            v.y = fmaxf(v.y + b.y, 0.0f);
            v.z = fmaxf(v.z + b.z, 0.0f);
            v.w = fmaxf(v.w + b.w, 0.0f);
            ((float4*)(out + (size_t)m * C128))[c4] = v;
        }
    }
}

// ---------------- classifier: out[N,40] = h @ Wc[128,40] + bc ----------------
// Block: 96 threads = 3 waves (column tiles 0,16,32); col guard at 40.
__global__ void __launch_bounds__(96)
classifier_kernel(const float* __restrict__ H,
                  const float* __restrict__ Wc,
                  const float* __restrict__ bc,
                  float* __restrict__ out,
                  int nrows, int ncls) {
    __shared__ float sH[16][LDSP];    // input tile, reused as 16x48 out-stage

    const int m0  = blockIdx.x * 16;
    const int tid = threadIdx.x;

    for (int idx = tid; idx < 16 * 32; idx += 96) {
        int row = idx >> 5;
        int c4  = idx & 31;
        int m = m0 + row;
        float4 h = {0.f, 0.f, 0.f, 0.f};
        if (m < nrows) h = ((const float4*)(H + (size_t)m * C128))[c4];
        ((float4*)&sH[row][0])[c4] = h;
    }
    __syncthreads();

    const int wave = tid >> 5;
    const int lane = tid & 31;
    const int n    = lane & 15;
    const int half = lane >> 4;
    const int col  = wave * 16 + n;      // 0..47
    const bool colok = (col < ncls);

    v8f acc = {};
    for (int k0 = 0; k0 < C128; k0 += 4) {
        const int k = k0 + 2 * half;
        v2f a, b;
        a.x = sH[n][k];     a.y = sH[n][k + 1];
        b.x = colok ? Wc[(size_t)k * ncls + col] : 0.0f;
        b.y = colok ? Wc[(size_t)(k + 1) * ncls + col] : 0.0f;
        acc = __builtin_amdgcn_wmma_f32_16x16x4_f32(false, a, false, b,
                                                    (short)0, acc, false, false);
    }

    __syncthreads();
    #pragma unroll
    for (int i = 0; i < 8; ++i) {
        sH[i + 8 * half][col] = acc[i];
    }
    __syncthreads();

    // coalesced write-out of the 16 x ncls tile with bias fused
    for (int idx = tid; idx < 16 * ncls; idx += 96) {
        int row = idx / ncls;
        int c   = idx - row * ncls;
        int m = m0 + row;
        if (m < nrows) {
            out[(size_t)m * ncls + c] = sH[row][c] + bc[c];
        }
    }
}

extern "C" void kernel_launch(void* const* d_in, const int* in_sizes, int n_in,
                              void* d_out, int out_size, void* d_ws, size_t ws_size,
                              hipStream_t stream) {
    const float* x    = (const float*)d_in[0];
    const int*   ei   = (const int*)d_in[1];    // [2, E] flat: row0 = src, row1 = dst
    const float* W_l1 = (const float*)d_in[2];
    const float* W_r1 = (const float*)d_in[3];
    const float* b1   = (const float*)d_in[4];
    const float* W_l2 = (const float*)d_in[5];
    const float* W_r2 = (const float*)d_in[6];
    const float* b2   = (const float*)d_in[7];
    const float* W_c  = (const float*)d_in[8];
    const float* b_c  = (const float*)d_in[9];
    float* out = (float*)d_out;

    const int N = in_sizes[0] / C128;
    const int E = in_sizes[1] / 2;
    const int NCLS = in_sizes[9];               // b_c length = 40

    const int* src = ei;
    const int* dst = ei + E;

    // workspace carve-up (floats): deg | deg_inv | agg | h1 | h2
    float* ws      = (float*)d_ws;
    float* deg     = ws;
    float* deg_inv = deg + N;
    float* agg     = deg_inv + N;
    float* h1      = agg + (size_t)N * C128;
    float* h2      = h1  + (size_t)N * C128;

    const int rowTiles   = (N + 15) / 16;
    const int edgeBlocks = (E + 7) / 8;

    // degrees
    fill_zero_kernel<<<2048, 256, 0, stream>>>(deg, (long long)N);
    degree_kernel<<<(E + 255) / 256, 256, 0, stream>>>(dst, deg, E);
    deginv_kernel<<<(N + 255) / 256, 256, 0, stream>>>(deg, deg_inv, N);

    // ---- layer 1 ----
    fill_zero_kernel<<<4096, 256, 0, stream>>>(agg, (long long)N * C128);
    aggregate_kernel<<<edgeBlocks, 256, 0, stream>>>(x, src, dst, agg, E);
    sage_gemm_kernel<<<rowTiles, 256, 0, stream>>>(agg, deg_inv, x, W_l1, W_r1, b1, h1, N);

    // ---- layer 2 ----
    fill_zero_kernel<<<4096, 256, 0, stream>>>(agg, (long long)N * C128);
    aggregate_kernel<<<edgeBlocks, 256, 0, stream>>>(h1, src, dst, agg, E);
    sage_gemm_kernel<<<rowTiles, 256, 0, stream>>>(agg, deg_inv, h1, W_l2, W_r2, b2, h2, N);

    // ---- classifier ----
    classifier_kernel<<<rowTiles, 96, 0, stream>>>(h2, W_c, b_c, out, N, NCLS);
}